// LogEntmaxBisect_21096879358427
// MI455X (gfx1250) — compile-verified
//
#include <hip/hip_runtime.h>

// log(entmax_bisect(x, alpha=1.5, n_iter=100)) over dim=-1.
// x: (2, 1024, 32000) fp32. One block per row; the whole row lives in VGPRs
// (32 floats/thread x 1024 threads = 32768 >= 32000), so HBM is touched once
// for read and once for write (~524 MB total -> ~22.5 us roofline @ 23.3 TB/s).
// Bisection inner loop is written in clang ext-vector form so it lowers to
// packed v_pk_add/pk_max/pk_fma f32 (2 elements/instruction): ~48 packed VALU
// per thread per iteration instead of 96 scalar ops.

#define THREADS 1024
#define NWAVES  (THREADS / 32)     // 32 wave32 waves per block
#define DDIM    32000
#define NF4     (DDIM / 4)         // 8000 float4 per row
#define NITERS  100

// native clang vector (NOT HIP_vector_type): nontemporal builtins accept it
// and elementwise arithmetic lowers to v_pk_*_f32 packed ops.
typedef __attribute__((ext_vector_type(4))) float v4f;

__device__ __forceinline__ float wave_sum(float v) {
#pragma unroll
  for (int o = 16; o > 0; o >>= 1) v += __shfl_xor(v, o, 32);
  return v;
}

__device__ __forceinline__ float wave_max(float v) {
#pragma unroll
  for (int o = 16; o > 0; o >>= 1) v = fmaxf(v, __shfl_xor(v, o, 32));
  return v;
}

__global__ __launch_bounds__(THREADS, 1)
void logentmax_bisect_kernel(const float* __restrict__ X, float* __restrict__ Y) {
  __shared__ float s_part[NWAVES];
  __shared__ float s_bcast;

  const int tid  = threadIdx.x;
  const int wid  = tid >> 5;
  const int lane = tid & 31;

  const float* __restrict__ xrow = X + (size_t)blockIdx.x * DDIM;
  float* __restrict__       yrow = Y + (size_t)blockIdx.x * DDIM;

  const float NEG_INF = -__builtin_inff();
  const float AM1     = 0.5f;                    // alpha - 1

  // ---- Load the entire row into registers (scaled: Xs = x * (alpha-1)) ----
  // Streamed once -> non-temporal so the 262MB input doesn't thrash 192MB L2.
  v4f v[8];
#pragma unroll
  for (int j = 0; j < 8; ++j) {
    const int p4 = j * THREADS + tid;
    if (p4 < NF4) {
      v4f t = __builtin_nontemporal_load(&((const v4f*)xrow)[p4]);
      v[j] = t * AM1;                              // v_pk_mul_f32
    } else {
      // padding lanes: -inf contributes 0 to p() and never wins the max
      v[j] = (v4f)(NEG_INF);
    }
  }

  // ---- Row max: packed max scan, wave butterfly, wave 0 folds partials ----
  v4f m4 = v[0];
#pragma unroll
  for (int j = 1; j < 8; ++j)
    m4 = __builtin_elementwise_max(m4, v[j]);      // v_pk_max_num_f32
  float m = fmaxf(fmaxf(m4.x, m4.y), fmaxf(m4.z, m4.w));
  m = wave_max(m);
  if (lane == 0) s_part[wid] = m;
  __syncthreads();
  if (wid == 0) {
    float t = wave_max(s_part[lane]);
    if (lane == 0) s_bcast = t;
  }
  __syncthreads();
  const float mx = s_bcast;

  // local f-sum over this thread's 32 resident elements: sum max(Xs - tau, 0)^2
  // Fully packed: per v4f chunk = 2x pk_add + 2x pk_max + 2x pk_fma.
  auto rowsum = [&](float tau) -> float {
    const v4f tv   = (v4f)(tau);
    const v4f zero = (v4f)(0.0f);
    v4f acc = zero;
#pragma unroll
    for (int j = 0; j < 8; ++j) {
      v4f t = __builtin_elementwise_max(v[j] - tv, zero);
      acc = __builtin_elementwise_fma(t, t, acc);
    }
    return (acc.x + acc.y) + (acc.z + acc.w);
  };

  // Block-wide sum; every thread returns the identical (uniform) total.
  // Two split barriers; wave 0 does the cross-wave fold (1 ds_load/lane)
  // and broadcasts one scalar. WAR on s_part/s_bcast is covered: wave 0
  // reads s_part between the barriers, and the next iteration's writes
  // happen only after the second barrier.
  auto block_sum = [&](float local) -> float {
    float ws = wave_sum(local);
    if (lane == 0) s_part[wid] = ws;
    __syncthreads();
    if (wid == 0) {
      float t = wave_sum(s_part[lane]);
      if (lane == 0) s_bcast = t;
    }
    __syncthreads();
    return s_bcast;
  };

  // ---- Bisection (reference semantics) ----
  float tau_lo = mx - 1.0f;                        // gp(1)
  // (1/d)^(alpha-1) = sqrt(1/32000)
  float dm     = (mx - 0.005590169943749474f) - tau_lo;
  float f_lo   = block_sum(rowsum(tau_lo)) - 1.0f;

  float tau_m = tau_lo;
  float f_m   = f_lo;

#pragma unroll 1
  for (int it = 0; it < NITERS; ++it) {
    dm *= 0.5f;
    tau_m = tau_lo + dm;
    f_m = block_sum(rowsum(tau_m)) - 1.0f;
    const bool frozen = (tau_m == tau_lo);         // fp fixed point reached
    if (f_m * f_lo >= 0.0f) tau_lo = tau_m;
    // Once fl(tau_lo + dm) == tau_lo, every remaining reference iteration is
    // an exact no-op (rounding is monotone, dm only shrinks), so breaking here
    // is bit-identical to running all 100 iterations. Uniform across block.
    if (frozen) break;
  }

  // ---- Output: log(p_m / S) = 2*log(t) - log(S), -inf where t <= 0 ----
  const float S    = f_m + 1.0f;                   // sum(p_m) from last reduce
  const float logS = __logf(S);

#pragma unroll
  for (int j = 0; j < 8; ++j) {
    const int p4 = j * THREADS + tid;
    if (p4 < NF4) {
      v4f d = v[j] - (v4f)(tau_m);                 // v_pk_add_f32
      v4f o;
      o.x = (d.x > 0.0f) ? (2.0f * __logf(d.x) - logS) : NEG_INF;
      o.y = (d.y > 0.0f) ? (2.0f * __logf(d.y) - logS) : NEG_INF;
      o.z = (d.z > 0.0f) ? (2.0f * __logf(d.z) - logS) : NEG_INF;
      o.w = (d.w > 0.0f) ? (2.0f * __logf(d.w) - logS) : NEG_INF;
      __builtin_nontemporal_store(o, &((v4f*)yrow)[p4]);   // write-once stream
    }
  }
}

extern "C" void kernel_launch(void* const* d_in, const int* in_sizes, int n_in,
                              void* d_out, int out_size, void* d_ws, size_t ws_size,
                              hipStream_t stream) {
  (void)n_in; (void)out_size; (void)d_ws; (void)ws_size;
  const float* x = (const float*)d_in[0];
  float*       y = (float*)d_out;
  const int rows = in_sizes[0] / DDIM;             // 2 * 1024 = 2048 rows
  logentmax_bisect_kernel<<<rows, THREADS, 0, stream>>>(x, y);
}